// VectorQuantizerLinearSoft_5282809774149
// MI455X (gfx1250) — compile-verified
//
#include <hip/hip_runtime.h>

typedef __attribute__((ext_vector_type(2))) float v2f;
typedef __attribute__((ext_vector_type(8))) float v8f;
typedef unsigned int u32x4 __attribute__((ext_vector_type(4)));
typedef int i32x8 __attribute__((ext_vector_type(8)));
typedef int i32x4 __attribute__((ext_vector_type(4)));

#define Kc 1024
#define Dd 256
#define Bn 65536
#define LDA 260      // latents tile row stride (floats), 1040B: 16B aligned, bank-skewed
#define LDB 260      // embedding chunk row stride (matches TDM pad: 256dw + 4dw pad)
#define DSTR 1032    // dist/p tile row stride

// ws layout (floats): [0..1023] avg_prob sums, [1024] sse, [1025] min-dist sum, [1032..2055] e2
#define WS_E2 1032

// d_out offsets (floats), tuple order: q, vq_loss, entropy, hard_idx, hard_q, p, cluster
#define OFF_Q   0ull
#define OFF_VQ  16777216ull
#define OFF_ENT 16777217ull
#define OFF_HI  16777218ull
#define OFF_HQ  16842754ull
#define OFF_P   33619970ull
#define OFF_CM  100728834ull

// ---- Tensor Data Mover: DMA a 128x256 f32 tile of `emb` (row `baseRow`) into LDS.
// D# group1 enables padding: insert 4 dwords after every 256 dwords -> LDS row
// stride of 260 floats == LDB, reproducing the bank-skewed layout for free.
__device__ __forceinline__ void tdm_stage_emb(const float* emb, int baseRow,
                                              unsigned ldsByteOff) {
  unsigned long long ga =
      (unsigned long long)(uintptr_t)(emb + (size_t)baseRow * Dd);
  u32x4 g0 = {
      1u,                                   // count=1 (valid), user mode, no gather
      ldsByteOff,                           // lds_addr
      (unsigned)(ga & 0xFFFFFFFFull),       // global_addr[31:0]
      (unsigned)((ga >> 32) & 0x1FFFFFFull) // global_addr[56:32]
          | (2u << 30)                      // type=2 ("image")
  };
  i32x8 g1 = {
      (2 << 16)        // data_size = 4 bytes
          | (1 << 20)  // pad_enable
          | (7 << 22)  // pad_interval = 256 dwords
          | (3 << 25), // pad_amount  = 4 dwords
      (int)(256u << 16),   // tensor_dim0 = 256 (bits 63:48)
      (int)(1024u << 16),  // tensor_dim0 hi = 0, tensor_dim1 = 1024 (bits 95:80)
      (int)(256u << 16),   // tensor_dim1 hi = 0, tile_dim0 = 256 (bits 127:112)
      128,                 // tile_dim1 = 128, tile_dim2 = 0
      256,                 // tensor_dim0_stride = 256 (bits 191:160)
      0, 0
  };
  i32x4 z4 = {0, 0, 0, 0};
  i32x8 z8 = {0, 0, 0, 0, 0, 0, 0, 0};
  __builtin_amdgcn_tensor_load_to_lds(g0, g1, z4, z4, z8, 0);
}

__global__ void __launch_bounds__(256) vq_prep(const float* __restrict__ emb,
                                               float* __restrict__ ws) {
  const int tid = threadIdx.x;
  const int g = blockIdx.x * 256 + tid;
  if (g < 1032) ws[g] = 0.0f;                 // zero accumulators every launch
  const int lane = tid & 31;
  const int row = blockIdx.x * 8 + (tid >> 5); // one wave per embedding row
  if (row < Kc) {
    float s = 0.0f;
    for (int d = lane; d < Dd; d += 32) { float x = emb[row * Dd + d]; s += x * x; }
    for (int off = 16; off > 0; off >>= 1) s += __shfl_xor(s, off);
    if (lane == 0) ws[WS_E2 + row] = s;
  }
}

__global__ void __launch_bounds__(256) vq_main(const float* __restrict__ lat,
                                               const float* __restrict__ emb,
                                               float* __restrict__ ws,
                                               float* __restrict__ out) {
  extern __shared__ float sm[];
  float* sLat = sm;                    // 16*LDA
  float* sEmb = sLat + 16 * LDA;       // 128*LDB
  float* sP   = sEmb + 128 * LDB;      // 16*DSTR : dist, then softmax p
  float* sE2  = sP + 16 * DSTR;        // 1024
  __shared__ float sL2[16];

  const int tid  = threadIdx.x;
  const int lane = tid & 31;
  const int w    = tid >> 5;
  const int nl   = lane & 15;
  const int hi   = lane >> 4;          // 0: K pair {0,1} / M base 0 ; 1: {2,3} / M base 8
  const int koff = hi * 2;
  const int m0   = blockIdx.x * 16;
  const unsigned embLds = (unsigned)(uintptr_t)sEmb;

  // ---- stage latents tile + e2 ----
  for (int idx = tid; idx < 16 * 64; idx += 256) {
    int r = idx >> 6, c4 = (idx & 63) << 2;
    *(float4*)&sLat[r * LDA + c4] = *(const float4*)&lat[(size_t)(m0 + r) * Dd + c4];
  }
  for (int n = tid; n < Kc; n += 256) sE2[n] = ws[WS_E2 + n];
  __syncthreads();
  if (tid < 16) {
    float s = 0.0f;
    for (int d = 0; d < Dd; ++d) { float x = sLat[tid * LDA + d]; s += x * x; }
    sL2[tid] = s;
  }
  __syncthreads();

  // ---- GEMM1: dist[16][1024] = l2 + e2 - 2 * lat x embT ----
  for (int cb = 0; cb < 8; ++cb) {
    __syncthreads();
    const int base = cb * 128;
    if (w == 0) {                                  // one TDM DMA per chunk
      tdm_stage_emb(emb, base, embLds);
      __builtin_amdgcn_s_wait_tensorcnt(0);
    }
    __syncthreads();
    const int erow = w * 16 + nl;                  // chunk-local codebook row (= N)
    v8f c = {};
    for (int d0 = 0; d0 < Dd; d0 += 4) {
      v2f a = *(const v2f*)&sLat[nl * LDA + d0 + koff];
      v2f b = *(const v2f*)&sEmb[erow * LDB + d0 + koff];
      c = __builtin_amdgcn_wmma_f32_16x16x4_f32(false, a, false, b, (short)0, c,
                                                false, false);
    }
    const int n = base + erow;
    const float e2v = sE2[n];
#pragma unroll
    for (int v = 0; v < 8; ++v) {
      int m = v + 8 * hi;
      sP[m * DSTR + n] = sL2[m] + e2v - 2.0f * c[v];
    }
  }
  __syncthreads();

  // ---- per-row argmin/min, hard outputs, softmax, p streaming ----
  int* hardIdxOut = (int*)(out + OFF_HI);
  for (int rr = 0; rr < 2; ++rr) {
    const int m = w * 2 + rr;
    float dmin = 3.4e38f; int imin = 0;
    for (int n = lane; n < Kc; n += 32) {
      float d = sP[m * DSTR + n];
      if (d < dmin) { dmin = d; imin = n; }
    }
    for (int off = 16; off > 0; off >>= 1) {
      float od = __shfl_xor(dmin, off);
      int   oi = __shfl_xor(imin, off);
      if (od < dmin || (od == dmin && oi < imin)) { dmin = od; imin = oi; }
    }
    if (lane == 0) {
      hardIdxOut[m0 + m] = imin;
      atomicAdd(&ws[1025], dmin);                    // cluster metric sum
    }
    for (int col = lane; col < Dd; col += 32)
      out[OFF_HQ + (size_t)(m0 + m) * Dd + col] = emb[(size_t)imin * Dd + col];

    float s = 0.0f;
    for (int n = lane; n < Kc; n += 32) {
      float e = __expf(-10.0f * (sP[m * DSTR + n] - dmin));
      sP[m * DSTR + n] = e; s += e;
    }
    for (int off = 16; off > 0; off >>= 1) s += __shfl_xor(s, off);
    const float inv = 1.0f / s;
    for (int n = lane; n < Kc; n += 32) {
      float p = sP[m * DSTR + n] * inv;
      sP[m * DSTR + n] = p;
      out[OFF_P + (size_t)(m0 + m) * Kc + n] = p;
    }
  }
  __syncthreads();

  // ---- avg_probs accumulation ----
  for (int n = tid; n < Kc; n += 256) {
    float s = 0.0f;
#pragma unroll
    for (int m = 0; m < 16; ++m) s += sP[m * DSTR + n];
    atomicAdd(&ws[n], s);
  }

  // ---- GEMM2: quantized[16][256] = P x E ----
  v8f q0 = {}, q1 = {};
  const int dc0 = (w * 2) * 16, dc1 = (w * 2 + 1) * 16;
  for (int cb = 0; cb < 8; ++cb) {
    __syncthreads();
    const int base = cb * 128;
    if (w == 0) {
      tdm_stage_emb(emb, base, embLds);
      __builtin_amdgcn_s_wait_tensorcnt(0);
    }
    __syncthreads();
    for (int k0 = 0; k0 < 128; k0 += 4) {
      v2f a = *(const v2f*)&sP[nl * DSTR + base + k0 + koff];
      const int kr = k0 + koff;
      v2f b0 = { sEmb[kr * LDB + dc0 + nl], sEmb[(kr + 1) * LDB + dc0 + nl] };
      q0 = __builtin_amdgcn_wmma_f32_16x16x4_f32(false, a, false, b0, (short)0, q0,
                                                 false, false);
      v2f b1 = { sEmb[kr * LDB + dc1 + nl], sEmb[(kr + 1) * LDB + dc1 + nl] };
      q1 = __builtin_amdgcn_wmma_f32_16x16x4_f32(false, a, false, b1, (short)0, q1,
                                                 false, false);
    }
  }

  // ---- write quantized + sse for vq_loss ----
  float sse = 0.0f;
#pragma unroll
  for (int v = 0; v < 8; ++v) {
    const int m = v + 8 * hi;
    float q = q0[v];
    float l = sLat[m * LDA + dc0 + nl];
    out[OFF_Q + (size_t)(m0 + m) * Dd + dc0 + nl] = q;
    float d = q - l; sse += d * d;
    q = q1[v];
    l = sLat[m * LDA + dc1 + nl];
    out[OFF_Q + (size_t)(m0 + m) * Dd + dc1 + nl] = q;
    d = q - l; sse += d * d;
  }
  for (int off = 16; off > 0; off >>= 1) sse += __shfl_xor(sse, off);
  if (lane == 0) atomicAdd(&ws[1024], sse);
}

__global__ void __launch_bounds__(1024) vq_final(const float* __restrict__ ws,
                                                 float* __restrict__ out) {
  __shared__ float red[1024];
  const int t = threadIdx.x;
  float avg = ws[t] * (1.0f / 65536.0f);
  red[t] = -avg * logf(avg + 1e-10f);
  __syncthreads();
  for (int s = 512; s > 0; s >>= 1) {
    if (t < s) red[t] += red[t + s];
    __syncthreads();
  }
  if (t == 0) {
    out[OFF_ENT] = red[0];
    out[OFF_VQ]  = 1.25f * ws[1024] / (65536.0f * 256.0f);
    out[OFF_CM]  = ws[1025] * (1.0f / 65536.0f);
  }
}

extern "C" void kernel_launch(void* const* d_in, const int* in_sizes, int n_in,
                              void* d_out, int out_size, void* d_ws, size_t ws_size,
                              hipStream_t stream) {
  (void)in_sizes; (void)n_in; (void)out_size; (void)ws_size;
  const float* lat = (const float*)d_in[0];
  const float* emb = (const float*)d_in[1];
  float* out = (float*)d_out;
  float* ws  = (float*)d_ws;

  vq_prep<<<128, 256, 0, stream>>>(emb, ws);
  const size_t smem = (size_t)(16 * LDA + 128 * LDB + 16 * DSTR + 1024) * sizeof(float);
  vq_main<<<Bn / 16, 256, smem, stream>>>(lat, emb, ws, out);
  vq_final<<<1, 1024, 0, stream>>>(ws, out);
}